// BispectrumCalculator_37855841747316
// MI455X (gfx1250) — compile-verified
//
#include <hip/hip_runtime.h>
#include <math.h>

// ---------------------------------------------------------------------------
// Bispectrum for B=64 signals of N=512 (real input).
//   Stage 1: DFT as f32 GEMM  Y = X * W  using V_WMMA_F32_16X16X4_F32
//            (full fp32 precision on the matrix pipe; twiddles computed
//             in-register with exact (s*f) mod N phase reduction).
//   Stage 2: bandwidth-bound 134 MB triple-product write, spectrum in LDS,
//            float4 (b128) coalesced stores.
//   Stage 3: echo target to the tail of d_out (tuple return).
// ---------------------------------------------------------------------------

typedef float v2f __attribute__((ext_vector_type(2)));
typedef float v8f __attribute__((ext_vector_type(8)));

#define NFFT   512
#define HALF_N 256
#define ROWS_PER_BLOCK 4

// ---------------------------------------------------------------------------
// Stage 1: one wave32 per 16x16 output tile of Y.
//   A (16x4, f32):  lane&15 = M (batch row), half = lane>>4 selects K pair.
//   B (4x16, f32):  lane&15 = N (freq col), VGPR v + 2*half = K (sample).
//   C/D (16x16):    VGPR v, half -> M = v + 8*half, lane&15 = N.
// Y[b,f] = sum_s x[b,s] * e^{-2*pi*i*s*f/N}  ->  re: cos, im: -sin.
// ---------------------------------------------------------------------------
__global__ __launch_bounds__(32) void dft_wmma_kernel(const float* __restrict__ x,
                                                      float2* __restrict__ spec) {
  const int f0   = blockIdx.x * 16;  // frequency tile
  const int b0   = blockIdx.y * 16;  // batch tile
  const int lane = threadIdx.x;      // 0..31 (wave32)
  const int col  = lane & 15;
  const int half = lane >> 4;

  const int   f     = f0 + col;
  const float scale = 6.28318530717958647692f / (float)NFFT;
  const float* xrow = x + (size_t)(b0 + (lane & 15)) * NFFT;

  v8f accR = {0.f, 0.f, 0.f, 0.f, 0.f, 0.f, 0.f, 0.f};
  v8f accI = {0.f, 0.f, 0.f, 0.f, 0.f, 0.f, 0.f, 0.f};

  for (int kk = 0; kk < NFFT; kk += 4) {
    const int s0 = kk + 2 * half;          // this lane's two K samples: s0, s0+1

    // A fragment: X[b0+m][s0], X[b0+m][s0+1]  (8B-aligned pair load)
    const float2 av = *reinterpret_cast<const float2*>(xrow + s0);
    v2f a; a.x = av.x; a.y = av.y;

    // B fragments: W[s][f], exact phase reduction (s*f) mod N keeps sincos
    // argument in [0, 2*pi) regardless of s*f magnitude.
    const int p0 = (s0 * f) & (NFFT - 1);
    const int p1 = ((s0 + 1) * f) & (NFFT - 1);
    float sn0, cs0, sn1, cs1;
    __sincosf((float)p0 * scale, &sn0, &cs0);
    __sincosf((float)p1 * scale, &sn1, &cs1);
    v2f br; br.x = cs0;  br.y = cs1;   // real part of e^{-i t} = cos
    v2f bi; bi.x = -sn0; bi.y = -sn1;  // imag part of e^{-i t} = -sin

    // D = A*B + C   (args: neg_a, A, neg_b, B, c_mod, C, reuse_a, reuse_b)
    accR = __builtin_amdgcn_wmma_f32_16x16x4_f32(false, a, false, br, (short)0,
                                                 accR, false, false);
    accI = __builtin_amdgcn_wmma_f32_16x16x4_f32(false, a, false, bi, (short)0,
                                                 accI, false, false);
  }

#pragma unroll
  for (int v = 0; v < 8; ++v) {
    const int row = b0 + v + 8 * half;
    spec[(size_t)row * NFFT + f] = make_float2(accR[v], accI[v]);
  }
}

// ---------------------------------------------------------------------------
// Stage 2: Bx[i,j] = y[i]*conj(y[j])*y[(j-i) mod N], fftshifted both axes.
// out[b][0][a][c] = Re Bx[(a+256)&511][(c+256)&511], out[b][1][a][c] = Im.
// One block: one batch, ROWS_PER_BLOCK shifted rows; spectrum lives in LDS.
// 128 threads x float4 real + float4 imag per row -> coalesced b128 stores.
// ---------------------------------------------------------------------------
__global__ __launch_bounds__(128) void bispec_kernel(const float2* __restrict__ spec,
                                                     float* __restrict__ out) {
  __shared__ float syr[NFFT];
  __shared__ float syi[NFFT];

  const int b  = blockIdx.y;
  const int a0 = blockIdx.x * ROWS_PER_BLOCK;

  const float2* ybase = spec + (size_t)b * NFFT;
  for (int t = threadIdx.x; t < NFFT; t += 128) {
    const float2 v = ybase[t];
    syr[t] = v.x;
    syi[t] = v.y;
  }
  __syncthreads();

  float* outb = out + (size_t)b * 2 * NFFT * NFFT;
  const int c0 = threadIdx.x * 4;

#pragma unroll
  for (int r = 0; r < ROWS_PER_BLOCK; ++r) {
    const int a = a0 + r;
    const int i = (a + HALF_N) & (NFFT - 1);
    const float yir = syr[i], yii = syi[i];

    float4 oR, oI;
    float* pr = reinterpret_cast<float*>(&oR);
    float* pi = reinterpret_cast<float*>(&oI);
#pragma unroll
    for (int t = 0; t < 4; ++t) {
      const int c = c0 + t;
      const int j = (c + HALF_N) & (NFFT - 1);
      const int k = (j - i) & (NFFT - 1);
      const float yjr = syr[j], yji = syi[j];
      const float ykr = syr[k], yki = syi[k];
      // t1 = y_i * conj(y_j)
      const float tr = yir * yjr + yii * yji;
      const float ti = yii * yjr - yir * yji;
      // Bx = t1 * y_k
      pr[t] = tr * ykr - ti * yki;
      pi[t] = tr * yki + ti * ykr;
    }
    *reinterpret_cast<float4*>(outb + (size_t)a * NFFT + c0) = oR;
    *reinterpret_cast<float4*>(outb + (size_t)NFFT * NFFT + (size_t)a * NFFT + c0) = oI;
  }
}

// ---------------------------------------------------------------------------
// Stage 3: tuple return -> append target after the bispectrum block.
// ---------------------------------------------------------------------------
__global__ void copy_target_kernel(const float* __restrict__ tgt,
                                   float* __restrict__ dst, int n) {
  const int idx = blockIdx.x * blockDim.x + threadIdx.x;
  if (idx < n) dst[idx] = tgt[idx];
}

extern "C" void kernel_launch(void* const* d_in, const int* in_sizes, int n_in,
                              void* d_out, int out_size, void* d_ws, size_t ws_size,
                              hipStream_t stream) {
  const float* target = (const float*)d_in[0];
  float*       out    = (float*)d_out;
  float2*      spec   = (float2*)d_ws;          // 64*512 complex = 256 KB

  const int B = in_sizes[0] / NFFT;             // 64

  // Stage 1: DFT via f32 WMMA. One wave per 16x16 tile of Y.
  dim3 g1(NFFT / 16, B / 16);
  dft_wmma_kernel<<<g1, 32, 0, stream>>>(target, spec);

  // Stage 2: 134 MB bandwidth-bound bispectrum write.
  dim3 g2(NFFT / ROWS_PER_BLOCK, B);
  bispec_kernel<<<g2, 128, 0, stream>>>(spec, out);

  // Stage 3: echo target (second tuple element).
  const int n = B * NFFT;
  copy_target_kernel<<<(n + 255) / 256, 256, 0, stream>>>(
      target, out + (size_t)B * 2 * NFFT * NFFT, n);
}